// FakeStyleGraph_59287728554153
// MI455X (gfx1250) — compile-verified
//
#include <hip/hip_runtime.h>
#include <hip/hip_bf16.h>

#define VV 10000
#define DD 300
#define BB 32
#define LL 200
#define KK 32
#define CHUNK 8
#define NCHUNK (LL / CHUNK)   // 25 -> 800 blocks
#define NTHREADS 320          // 10 wave32s; one lane per d (300 used)

typedef __attribute__((ext_vector_type(2))) float v2f;
typedef __attribute__((ext_vector_type(8))) float v8f;

__global__ void zero_rep_kernel(float* __restrict__ rep, int n) {
  int i = blockIdx.x * blockDim.x + threadIdx.x;
  if (i < n) rep[i] = 0.0f;
}

__global__ __launch_bounds__(NTHREADS) void gather_max_acc_kernel(
    const int* __restrict__ nodes, const int* __restrict__ neighbors,
    const float* __restrict__ embedding, const float* __restrict__ adjacency,
    const float* __restrict__ aggregate_rate, const int* __restrict__ pad_id_p,
    float* __restrict__ rep) {
  __shared__ float s_h[KK];
  __shared__ int   s_nbr[KK];

  const int b   = blockIdx.x / NCHUNK;
  const int l0  = (blockIdx.x % NCHUNK) * CHUNK;
  const int tid = threadIdx.x;
  const int pad = *pad_id_p;

  float acc = 0.0f;

  for (int li = 0; li < CHUNK; ++li) {
    const int l    = l0 + li;
    const int node = nodes[b * LL + l];         // block-uniform
    if (node == pad) continue;                  // uniform branch: exact (rate=ori=0)

    if (tid < KK) {
      const int nb = neighbors[(size_t)(b * LL + l) * KK + tid];
      s_nbr[tid] = nb;
      s_h[tid]   = adjacency[(size_t)node * VV + nb];
    }
    if (tid == 0 && li + 1 < CHUNK) {
      // gfx1250 global_prefetch_b8: warm next l's neighbor indices
      __builtin_prefetch(&neighbors[(size_t)(b * LL + l + 1) * KK], 0, 0);
    }
    __syncthreads();

    if (tid < DD) {
      const float rate = aggregate_rate[node];  // wave-uniform
      float m = -__builtin_inff();
      #pragma unroll 8
      for (int k = 0; k < KK; ++k) {
        // s_nbr[k] is wave-uniform -> scalar base; lanes read coalesced 128B
        m = fmaxf(m, s_h[k] * embedding[(size_t)s_nbr[k] * DD + tid]);
      }
      acc += rate * m + (1.0f - rate) * embedding[(size_t)node * DD + tid];
    }
    __syncthreads();
  }

  if (tid < DD) atomicAdd(&rep[b * DD + tid], acc);
}

// out[32,2] = rep[32,300] @ w[300,2] + bias, via V_WMMA_F32_16X16X4_F32.
// A (16x4 f32): lanes 0-15 row M=lane, VGPR0/1 = K{0,1}; lanes 16-31 = K{2,3}.
// C/D (16x16 f32): VGPR j -> (M=j, N=lane) for lanes 0-15, (M=j+8, N=lane-16).
__global__ __launch_bounds__(32) void out_wmma_kernel(
    const float* __restrict__ rep, const float* __restrict__ w,
    const float* __restrict__ bias, float* __restrict__ out) {
  const int lane = threadIdx.x;   // 0..31, EXEC all ones at WMMA
  const int half = lane >> 4;     // selects K pair {0,1} vs {2,3}
  const int lid  = lane & 15;
  const float b0 = bias[0];
  const float b1 = bias[1];

  #pragma unroll
  for (int tile = 0; tile < 2; ++tile) {       // M = 32 -> two 16-row tiles
    v8f c = {};
    const int row = tile * 16 + lid;
    for (int k0 = 0; k0 < DD; k0 += 4) {       // 75 WMMA steps
      const int ka = k0 + 2 * half;
      v2f a, bv;
      a.x = rep[row * DD + ka + 0];
      a.y = rep[row * DD + ka + 1];
      // B (4x16): column N = lid; only N=0,1 are real, rest zero-padded
      const float wm = (lid < 2) ? 1.0f : 0.0f;
      const int   wn = (lid < 2) ? lid : 0;
      bv.x = wm * w[(ka + 0) * 2 + wn];
      bv.y = wm * w[(ka + 1) * 2 + wn];
      c = __builtin_amdgcn_wmma_f32_16x16x4_f32(
          /*neg_a=*/false, a, /*neg_b=*/false, bv,
          /*c_mod=*/(short)0, c, /*reuse_a=*/false, /*reuse_b=*/false);
    }
    if (lid < 2) {
      const float bb = (lid == 0) ? b0 : b1;
      #pragma unroll
      for (int j = 0; j < 8; ++j) {
        const int m = tile * 16 + half * 8 + j;
        out[m * 2 + lid] = c[j] + bb;
      }
    }
  }
}

extern "C" void kernel_launch(void* const* d_in, const int* in_sizes, int n_in,
                              void* d_out, int out_size, void* d_ws, size_t ws_size,
                              hipStream_t stream) {
  const int*   nodes          = (const int*)d_in[0];
  const int*   neighbors      = (const int*)d_in[1];
  const float* embedding      = (const float*)d_in[2];
  const float* adjacency      = (const float*)d_in[3];
  const float* aggregate_rate = (const float*)d_in[4];
  const float* w_last         = (const float*)d_in[5];
  const float* b_last         = (const float*)d_in[6];
  const int*   pad_id         = (const int*)d_in[7];

  float* out = (float*)d_out;
  float* rep = (float*)d_ws;   // BB*DD floats = 38400 bytes of scratch

  zero_rep_kernel<<<(BB * DD + 255) / 256, 256, 0, stream>>>(rep, BB * DD);

  gather_max_acc_kernel<<<BB * NCHUNK, NTHREADS, 0, stream>>>(
      nodes, neighbors, embedding, adjacency, aggregate_rate, pad_id, rep);

  out_wmma_kernel<<<1, 32, 0, stream>>>(rep, w_last, b_last, out);
}